// PercentileDiscretizer_67542655697717
// MI455X (gfx1250) — compile-verified
//
#include <hip/hip_runtime.h>
#include <stdint.h>

// ---- problem constants (match reference) ----
#define N_FEATURE   100000
#define N_BIN       16
#define DELIMS      (N_BIN + 1)          // 17
#define OUT_BITS    22
#define OUT_MASK    ((1u << OUT_BITS) - 1u)

// ---- two-level search: LDS sample table ----
#define SAMPLE_STRIDE 16
#define N_SAMPLE ((N_FEATURE + SAMPLE_STRIDE - 1) / SAMPLE_STRIDE)   // 6250 -> 50000 B LDS

typedef float f4 __attribute__((ext_vector_type(4)));
struct __attribute__((packed, aligned(4))) PackedF4 { f4 v; };       // dword-aligned b128 view

__global__ __launch_bounds__(256) void PercentileDiscretizer_kernel(
    const long long* __restrict__ ids,
    const long long* __restrict__ keys,
    const float*     __restrict__ vals,
    const long long* __restrict__ hash_keys,
    const long long* __restrict__ hash_values,
    const long long* __restrict__ bin_ids,
    const float*     __restrict__ bin_values,
    const long long* __restrict__ feature_offsets,
    float* __restrict__ out_ids,
    float* __restrict__ out_keys,
    float* __restrict__ out_vals,
    int nnz)
{
    __shared__ long long s_sample[N_SAMPLE];

    // ---- cooperative LDS fill via CDNA5 async global->LDS copy (GVS mode) ----
    // global data at saddr + vaddr offset goes straight to LDS at vdst; tracked on ASYNCcnt, no VGPR round-trip.
    for (unsigned t = threadIdx.x; t < N_SAMPLE; t += 256u) {
        unsigned goff = (unsigned)(t * SAMPLE_STRIDE) * 8u;          // byte offset into hash_keys
        unsigned lds  = (unsigned)(size_t)(&s_sample[t]);            // LDS byte address
        asm volatile("global_load_async_to_lds_b64 %0, %1, %2"
                     :: "v"(lds), "v"(goff), "s"(hash_keys)
                     : "memory");
    }
    asm volatile("s_wait_asynccnt 0" ::: "memory");                  // my async copies landed
    __syncthreads();                                                 // everyone's landed

    const int stride = (int)(gridDim.x * 256u);
    for (int i = (int)(blockIdx.x * 256u + threadIdx.x); i < nnz; i += stride) {
        // Streaming data: non-temporal so the 160MB of stream traffic does not
        // evict the L2-resident calibration tables (~22MB, ~40x reuse).
        const long long key = __builtin_nontemporal_load(&keys[i]);
        const float     val = __builtin_nontemporal_load(&vals[i]);
        const long long id  = __builtin_nontemporal_load(&ids[i]);
        __builtin_nontemporal_store((float)id, &out_ids[i]);

        // ---- level 1: lower_bound over LDS sample table (~13 LDS probes) ----
        unsigned lo = 0, hi = N_SAMPLE;
        while (lo < hi) {
            unsigned mid = (lo + hi) >> 1;
            if (s_sample[mid] < key) lo = mid + 1; else hi = mid;
        }
        const unsigned t = lo;

        // ---- level 2: lower_bound inside a <=16-entry global window (<=5 probes) ----
        // S[t-1] < key  =>  pos >= (t-1)*16 + 1 ;  S[t] >= key  =>  pos <= t*16
        unsigned wlo = (t > 0)        ? (t - 1) * SAMPLE_STRIDE + 1 : 0;
        unsigned whi = (t < N_SAMPLE) ? t * SAMPLE_STRIDE + 1       : N_FEATURE;
        if (whi > N_FEATURE) whi = N_FEATURE;
        while (wlo < whi) {
            unsigned mid = (wlo + whi) >> 1;
            if (hash_keys[mid] < key) wlo = mid + 1; else whi = mid;
        }
        const unsigned pos   = wlo;
        const unsigned pos_c = (pos < N_FEATURE - 1u) ? pos : (N_FEATURE - 1u);
        const bool found = (hash_keys[pos_c] == key);

        float okey, oval;
        if (found) {
            const long long j = hash_values[pos_c];
            const float* __restrict__ drow = bin_values + (size_t)j * DELIMS;
            __builtin_prefetch(drow, 0, 0);                          // global_prefetch_b8
            // bucketize: count of (val >= delim); 4 x b128 + 1 x b32 independent gathers
            const PackedF4* __restrict__ p = (const PackedF4*)drow;
            const f4 d0 = p[0].v, d1 = p[1].v, d2 = p[2].v, d3 = p[3].v;
            const float d16 = drow[16];
            int cnt = 0;
            cnt += (val >= d0.x) + (val >= d0.y) + (val >= d0.z) + (val >= d0.w);
            cnt += (val >= d1.x) + (val >= d1.y) + (val >= d1.z) + (val >= d1.w);
            cnt += (val >= d2.x) + (val >= d2.y) + (val >= d2.z) + (val >= d2.w);
            cnt += (val >= d3.x) + (val >= d3.y) + (val >= d3.z) + (val >= d3.w);
            cnt += (val >= d16);
            int bin = cnt - 1;
            bin = bin < 0 ? 0 : (bin > N_BIN - 1 ? N_BIN - 1 : bin);
            const long long off = feature_offsets[j];
            const long long dk  = bin_ids[off + bin];
            okey = (float)dk;
            oval = 1.0f;
        } else {
            okey = (float)((unsigned)key & OUT_MASK);
            oval = val;
        }
        __builtin_nontemporal_store(okey, &out_keys[i]);
        __builtin_nontemporal_store(oval, &out_vals[i]);
    }
}

extern "C" void kernel_launch(void* const* d_in, const int* in_sizes, int n_in,
                              void* d_out, int out_size, void* d_ws, size_t ws_size,
                              hipStream_t stream) {
    (void)n_in; (void)out_size; (void)d_ws; (void)ws_size;
    const long long* ids             = (const long long*)d_in[0];
    const long long* keys            = (const long long*)d_in[1];
    const float*     vals            = (const float*)d_in[2];
    const long long* hash_keys       = (const long long*)d_in[3];
    const long long* hash_values     = (const long long*)d_in[4];
    const long long* bin_ids         = (const long long*)d_in[5];
    const float*     bin_values      = (const float*)d_in[6];
    const long long* feature_offsets = (const long long*)d_in[7];

    const int nnz = in_sizes[0];
    float* out = (float*)d_out;
    float* out_ids  = out;
    float* out_keys = out + nnz;
    float* out_vals = out + 2 * (size_t)nnz;

    const int threads = 256;                       // 8 wave32 waves per workgroup
    int blocks = (nnz + threads - 1) / threads;
    if (blocks > 1024) blocks = 1024;              // grid-stride; amortize 50KB LDS table fill

    PercentileDiscretizer_kernel<<<blocks, threads, 0, stream>>>(
        ids, keys, vals, hash_keys, hash_values, bin_ids, bin_values,
        feature_offsets, out_ids, out_keys, out_vals, nnz);
}